// HyperedgeAggregator_29643864277335
// MI455X (gfx1250) — compile-verified
//
#include <hip/hip_runtime.h>
#include <stdint.h>

#define EMBED_DIM   256
#define NUM_EDGES   16384
#define TOTAL_MEM   524288
#define IDX_CHUNK   256     // indices staged per outer iteration
#define CHUNK       16      // member rows per pipeline stage (per wave slice)

// One block per segment. 8 waves; wave w owns dims [32w, 32w+32).
// Member-row slices are pulled into per-wave LDS regions with
// global_load_async_to_lds_b128, double-buffered via ASYNCcnt.
__global__ __launch_bounds__(256) void hyperedge_mean_kernel(
    const float* __restrict__ emb,    // [NUM_NODES, 256] f32
    const int*   __restrict__ nidx,   // [TOTAL] member node indices
    const int*   __restrict__ segid,  // [TOTAL] sorted segment ids
    float*       __restrict__ out)    // [NUM_EDGES, 256] f32
{
    __shared__ int   sidx[IDX_CHUNK];
    __shared__ float gbuf[8][2][CHUNK][32];   // 32 KB: per-wave double buffer

    const int seg  = blockIdx.x;
    const int lane = threadIdx.x & 31;
    const int w    = threadIdx.x >> 5;

    // lower_bound(seg) and lower_bound(seg+1) over sorted segid (uniform -> scalar)
    int lo = 0, hi = TOTAL_MEM;
    while (lo < hi) { int mid = (lo + hi) >> 1; if (segid[mid] < seg) lo = mid + 1; else hi = mid; }
    const int start = lo;
    hi = TOTAL_MEM;
    while (lo < hi) { int mid = (lo + hi) >> 1; if (segid[mid] < seg + 1) lo = mid + 1; else hi = mid; }
    const int cnt = lo - start;

    float acc = 0.0f;

    // LDS byte address of this wave's staging region (low 32 bits of flat LDS addr == DS address)
    const unsigned ldsbase = (unsigned)(uintptr_t)&gbuf[w][0][0][0];
    // Uniform SGPR base: embedding table + this wave's 128B dim-slice offset
    const unsigned long long saddr =
        (unsigned long long)(uintptr_t)emb + (unsigned long long)(w * 128);
    const int      j4  = lane >> 3;            // member-within-group 0..3
    const unsigned b16 = (unsigned)(lane & 7) << 4;  // 16B sub-slice 0..112

    for (int base = 0; base < cnt; base += IDX_CHUNK) {
        int m = cnt - base; if (m > IDX_CHUNK) m = IDX_CHUNK;

        __syncthreads();   // protect sidx reuse across outer iterations
        for (int t = threadIdx.x; t < m; t += 256) sidx[t] = nidx[start + base + t];
        __syncthreads();

        const int nchunk = (m + CHUNK - 1) / CHUNK;

        // Issue one 16-member chunk: 4 x b128 async loads (8 lanes x 16B x 4 members each)
        auto issue = [&](int c, int stage) {
            const int cb = c * CHUNK;
            #pragma unroll
            for (int g = 0; g < 4; ++g) {
                int mj = cb + g * 4 + j4;
                if (mj > m - 1) mj = m - 1;   // pad tail with a valid row (never consumed)
                unsigned voff = (unsigned)sidx[mj] * 1024u + b16;           // per-lane i32 offset
                unsigned lds  = ldsbase + (unsigned)stage * (CHUNK * 128u)
                              + (unsigned)(g * 4 + j4) * 128u + b16;
                asm volatile("global_load_async_to_lds_b128 %0, %1, %2"
                             :: "v"(lds), "v"(voff), "s"(saddr)
                             : "memory");
            }
        };

        if (nchunk > 0) issue(0, 0);
        if (nchunk > 1) issue(1, 1);

        for (int c = 0; c < nchunk; ++c) {
            if (c + 1 < nchunk) {
                // chunks c and c+1 in flight (<=8); in-order completion: <=4 means chunk c done
                asm volatile("s_wait_asynccnt 4" ::: "memory");
            } else {
                asm volatile("s_wait_asynccnt 0" ::: "memory");
            }
            const int stage = c & 1;
            const int cb    = c * CHUNK;
            #pragma unroll
            for (int j = 0; j < CHUNK; ++j) {
                if (cb + j < m) acc += gbuf[w][stage][j][lane];
            }
            if (c + 2 < nchunk) issue(c + 2, stage);  // refill the stage just consumed
        }
    }

    const float inv = 1.0f / (float)(cnt > 0 ? cnt : 1);
    out[(size_t)seg * EMBED_DIM + w * 32 + lane] = acc * inv;
}

extern "C" void kernel_launch(void* const* d_in, const int* in_sizes, int n_in,
                              void* d_out, int out_size, void* d_ws, size_t ws_size,
                              hipStream_t stream) {
    const float* emb   = (const float*)d_in[0];
    const int*   nidx  = (const int*)d_in[1];
    const int*   segid = (const int*)d_in[2];
    float*       out   = (float*)d_out;
    (void)in_sizes; (void)n_in; (void)out_size; (void)d_ws; (void)ws_size;

    hipLaunchKernelGGL(hyperedge_mean_kernel,
                       dim3(NUM_EDGES), dim3(256), 0, stream,
                       emb, nidx, segid, out);
}